// DEMA_59141699665995
// MI455X (gfx1250) — compile-verified
//
#include <hip/hip_runtime.h>
#include <hip/hip_bf16.h>

// DEMA (Holt double exponential smoothing) scan over T.
// x: (B=64, T=2048, D=512) fp32. 32768 independent sequences of length 2048.
// Memory-bound: 512 MB total traffic -> ~22us floor at 23.3 TB/s.
// Strategy: 1 thread per (b,f) sequence, 32-deep async-to-LDS prefetch ring
// (gfx1250 GLOBAL_LOAD_ASYNC_TO_LDS_B32 / S_WAIT_ASYNCCNT), NT stores for out.
// In-flight reads = 32768 threads * 4B * PF = 4 MB at PF=32 (ASYNCcnt max 63).

#define B_LEN 64
#define T_LEN 2048
#define D_LEN 512
#define BD    64               // threads per block (2 wave32) -> 512 blocks
#define PF    32               // prefetch ring depth (power of 2, divides T_LEN)

// Wait until <= N async-to-LDS ops outstanding. asm volatile + memory clobber
// so the compiler cannot reorder the subsequent LDS reads above the wait.
#define WAIT_ASYNC(N) asm volatile("s_wait_asynccnt %0" :: "i"(N) : "memory")

typedef __attribute__((address_space(1))) int  gint_t;   // global (AS1)
typedef __attribute__((address_space(3))) int  lint_t;   // LDS (AS3)

__device__ __forceinline__ void async_ld_f32(const float* gp, float* lp) {
#if __has_builtin(__builtin_amdgcn_global_load_async_to_lds_b32)
    __builtin_amdgcn_global_load_async_to_lds_b32(
        (gint_t*)gp, (lint_t*)lp, /*offset=*/0, /*cpol=*/0);
#else
    unsigned loff = (unsigned)(unsigned long long)
        (__attribute__((address_space(3))) char*)lp;
    asm volatile("global_load_async_to_lds_b32 %0, %1, off"
                 :: "v"(loff), "v"(gp) : "memory");
#endif
}

__global__ __launch_bounds__(BD) void dema_scan_kernel(
    const float* __restrict__ x,
    const float* __restrict__ alpha_p,
    const float* __restrict__ beta_p,
    float* __restrict__ out)
{
    __shared__ float buf[PF * BD];   // 8 KB ring: PF stages x BD lanes

    const int tid   = threadIdx.x;
    const int b     = blockIdx.x >> 3;                 // 8 f-chunks per batch
    const int f     = ((blockIdx.x & 7) << 6) + tid;   // chunk*64 + lane
    const size_t base = ((size_t)b * T_LEN) * D_LEN + (size_t)f;
    const float* xp = x + base;                        // stride D_LEN per t
    float*       op = out + base;

    const float alpha = alpha_p[0];                    // uniform -> s_load
    const float beta  = beta_p[0];
    const float oma   = 1.0f - alpha;
    const float omb   = 1.0f - beta;

    // ---- prologue: fill the async pipeline with t = 0..PF-1
    #pragma unroll
    for (int k = 0; k < PF; ++k)
        async_ld_f32(xp + (size_t)k * D_LEN, &buf[k * BD + tid]);

    float s, btr;

    // ---- t = 0 : s0 = x0
    WAIT_ASYNC(PF - 1);                 // oldest stage (t=0) has landed
    {
        float v0 = buf[0 * BD + tid];
        s = v0;
        __builtin_nontemporal_store(s, op);
        async_ld_f32(xp + (size_t)PF * D_LEN, &buf[0 * BD + tid]);
    }

    // ---- t = 1 : b0 = x1 - s0, then one recurrence step
    WAIT_ASYNC(PF - 1);
    {
        float v1 = buf[1 * BD + tid];
        btr = v1 - s;
        float sp = s + btr;
        float sn = fmaf(alpha, v1, oma * sp);
        btr = fmaf(beta, sn - s, omb * btr);
        s = sn;
        __builtin_nontemporal_store(s, op + D_LEN);
        async_ld_f32(xp + (size_t)(PF + 1) * D_LEN, &buf[1 * BD + tid]);
    }

    // ---- steady state: consume stage t, re-arm it with t+PF
    #pragma unroll 8
    for (int t = 2; t < T_LEN - PF; ++t) {
        WAIT_ASYNC(PF - 1);             // in-order completion => load t done
        const int st = t & (PF - 1);
        float v  = buf[st * BD + tid];
        float sp = s + btr;
        float sn = fmaf(alpha, v, oma * sp);
        btr = fmaf(beta, sn - s, omb * btr);
        s = sn;
        __builtin_nontemporal_store(s, op + (size_t)t * D_LEN);
        async_ld_f32(xp + (size_t)(t + PF) * D_LEN, &buf[st * BD + tid]);
    }

    // ---- drain: all remaining loads already issued; wait once, consume PF
    WAIT_ASYNC(0);
    #pragma unroll
    for (int t = T_LEN - PF; t < T_LEN; ++t) {
        float v  = buf[(t & (PF - 1)) * BD + tid];
        float sp = s + btr;
        float sn = fmaf(alpha, v, oma * sp);
        btr = fmaf(beta, sn - s, omb * btr);
        s = sn;
        __builtin_nontemporal_store(s, op + (size_t)t * D_LEN);
    }
}

extern "C" void kernel_launch(void* const* d_in, const int* in_sizes, int n_in,
                              void* d_out, int out_size, void* d_ws, size_t ws_size,
                              hipStream_t stream) {
    const float* x     = (const float*)d_in[0];
    const float* alpha = (const float*)d_in[1];
    const float* beta  = (const float*)d_in[2];
    float*       out   = (float*)d_out;

    dim3 grid(B_LEN * (D_LEN / BD));   // 64 * 8 = 512 blocks
    dim3 block(BD);                    // 64 threads = 2 wave32
    dema_scan_kernel<<<grid, block, 0, stream>>>(x, alpha, beta, out);
}